// SIM_64166811402423
// MI455X (gfx1250) — compile-verified
//
#include <hip/hip_runtime.h>
#include <hip/hip_bf16.h>
#include <math.h>

#define BATCH 16384
#define SEQ   50
#define TOPK  5
#define HD    32

typedef __attribute__((ext_vector_type(2))) float v2f;
typedef __attribute__((ext_vector_type(8))) float v8f;

__device__ __forceinline__ float wave_sum(float v) {
    // wave32 full butterfly: result broadcast to all lanes
    for (int off = 16; off > 0; off >>= 1) v += __shfl_xor(v, off, 32);
    return v;
}

// ---------------------------------------------------------------------------
// Kernel: pad au_W1 (129x36 -> 132x48) and l2_W (80x40 -> 80x48) with zeros
// ---------------------------------------------------------------------------
__global__ void prep_weights(const float* __restrict__ W1, const float* __restrict__ W2,
                             float* __restrict__ W1P, float* __restrict__ W2P) {
    int t = blockIdx.x * blockDim.x + threadIdx.x;
    if (t < 132 * 48) {
        int k = t / 48, n = t % 48;
        W1P[t] = (k < 129 && n < 36) ? W1[k * 36 + n] : 0.f;
    }
    if (t < 80 * 48) {
        int k = t / 48, n = t % 48;
        W2P[t] = (n < 40) ? W2[k * 40 + n] : 0.f;
    }
}

// ---------------------------------------------------------------------------
// Kernel: one wave32 per batch row.
// Gathers embeddings, cosine sim over 50 history items, streaming top-5,
// writes attention-input rows X (B*5 x 132, cols: hw[64], cross, item_cat[64],
// zero-pad[3]) and top-k rate embeddings TR (B*5 x 32).
// ---------------------------------------------------------------------------
__global__ __launch_bounds__(256) void build_x(
    const int* __restrict__ iid, const int* __restrict__ aid,
    const int* __restrict__ hist_iid, const int* __restrict__ hist_aid,
    const int* __restrict__ hist_rate,
    const float* __restrict__ item_tab, const float* __restrict__ cate_tab,
    const float* __restrict__ rate_tab,
    float* __restrict__ X, float* __restrict__ TR) {
    const int wave = threadIdx.x >> 5;
    const int lane = threadIdx.x & 31;
    const int b = blockIdx.x * 8 + wave;

    const float ic0 = item_tab[(size_t)iid[b] * HD + lane];
    const float ic1 = cate_tab[(size_t)aid[b] * HD + lane];
    const float n1 = sqrtf(wave_sum(ic0 * ic0 + ic1 * ic1));

    float tv[TOPK];
    int   ti[TOPK];
#pragma unroll
    for (int j = 0; j < TOPK; ++j) { tv[j] = -1e30f; ti[j] = 0; }

    for (int s = 0; s < SEQ; ++s) {
        const float h0 = item_tab[(size_t)hist_iid[b * SEQ + s] * HD + lane];
        const float h1 = cate_tab[(size_t)hist_aid[b * SEQ + s] * HD + lane];
        const float dot = wave_sum(ic0 * h0 + ic1 * h1);
        const float n2 = sqrtf(wave_sum(h0 * h0 + h1 * h1));
        const float sim = dot / fmaxf(n1 * n2, 1e-8f);
        // stable insert (strict >) preserves lower-index-first on ties
        int pos = TOPK;
#pragma unroll
        for (int j = TOPK - 1; j >= 0; --j) { if (sim > tv[j]) pos = j; }
        if (pos < TOPK) {
#pragma unroll
            for (int j = TOPK - 1; j > 0; --j) {
                if (j > pos) { tv[j] = tv[j - 1]; ti[j] = ti[j - 1]; }
            }
            tv[pos] = sim; ti[pos] = s;
        }
    }

    const float wsum = tv[0] + tv[1] + tv[2] + tv[3] + tv[4] + 1e-8f;
#pragma unroll
    for (int k = 0; k < TOPK; ++k) {
        const int s = ti[k];
        const float wk = tv[k] / wsum;
        const float t0 = item_tab[(size_t)hist_iid[b * SEQ + s] * HD + lane];
        const float t1 = cate_tab[(size_t)hist_aid[b * SEQ + s] * HD + lane];
        const float hw0 = wk * t0, hw1 = wk * t1;
        const float cross = wave_sum(hw0 * ic0 + hw1 * ic1);
        const size_t ro = (size_t)(b * TOPK + k) * 132;
        X[ro + lane]       = hw0;
        X[ro + 32 + lane]  = hw1;
        X[ro + 65 + lane]  = ic0;
        X[ro + 97 + lane]  = ic1;
        if (lane == 0) X[ro + 64] = cross;
        if (lane < 3)  X[ro + 129 + lane] = 0.f;
        TR[(size_t)(b * TOPK + k) * HD + lane] =
            rate_tab[(size_t)hist_rate[b * SEQ + s] * HD + lane];
    }
}

// ---------------------------------------------------------------------------
// fp32 WMMA GEMM: C(M x Nlog) = A(M x Kp, lda) * B(Kp x Nlog, ldb) + bias.
// One wave handles a 16-row strip and ALL NT n-tiles: the A fragment is
// loaded once per k-step and reused across NT independent WMMA accumulator
// chains (A traffic /NT, better WMMA pipeline overlap). 8 waves per block.
// Fragment layouts per CDNA5 ISA (16x4 A / 4x16 B, fp32):
//   A: lane<16 holds A[m0+r][k], A[m0+r][k+1]; lane>=16 holds k+2,k+3
//   B: lane<16 holds B[k][n0+r], B[k+1][n0+r]; lane>=16 holds k+2,k+3 rows
// ---------------------------------------------------------------------------
template <int NT>
__global__ __launch_bounds__(256) void wmma_gemm_f32(
    const float* __restrict__ A, int lda,
    const float* __restrict__ Bm, int ldb,
    const float* __restrict__ bias,
    float* __restrict__ C, int ldc, int Nlog, int Kp) {
    const int wave = threadIdx.x >> 5;
    const int lane = threadIdx.x & 31;
    const int m0 = (blockIdx.x * 8 + wave) * 16;
    const int half = lane >> 4;
    const int r = lane & 15;

    v8f acc[NT];
#pragma unroll
    for (int t = 0; t < NT; ++t) {
        const int col = t * 16 + r;
        const float bval = (col < Nlog) ? bias[col] : 0.f;
#pragma unroll
        for (int i = 0; i < 8; ++i) acc[t][i] = bval;
    }

    for (int k = 0; k < Kp; k += 4) {
        const float* ap = A + (size_t)(m0 + r) * lda + k + 2 * half;
        v2f a; a.x = ap[0]; a.y = ap[1];
        const float* bp = Bm + (size_t)(k + 2 * half) * ldb + r;
#pragma unroll
        for (int t = 0; t < NT; ++t) {
            v2f bf; bf.x = bp[t * 16]; bf.y = bp[ldb + t * 16];
            acc[t] = __builtin_amdgcn_wmma_f32_16x16x4_f32(
                         false, a, false, bf, (short)0, acc[t], false, false);
        }
    }

    const int rowbase = m0 + 8 * half;
#pragma unroll
    for (int t = 0; t < NT; ++t) {
        const int col = t * 16 + r;
        if (col < Nlog) {
#pragma unroll
            for (int i = 0; i < 8; ++i)
                C[(size_t)(rowbase + i) * ldc + col] = acc[t][i];
        }
    }
}

// ---------------------------------------------------------------------------
// Per-channel sum & sum-of-squares over R x C (row-major), C <= 80.
// sums[0..C) = sum, sums[C..2C) = sum of squares. Must be pre-zeroed.
// ---------------------------------------------------------------------------
__global__ __launch_bounds__(256) void stats_kernel(
    const float* __restrict__ x, int R, int C, float* __restrict__ sums) {
    __shared__ float sbuf[160];
    for (int i = threadIdx.x; i < 2 * C; i += blockDim.x) sbuf[i] = 0.f;
    __syncthreads();
    const size_t total = (size_t)R * C;
    for (size_t idx = (size_t)blockIdx.x * blockDim.x + threadIdx.x;
         idx < total; idx += (size_t)gridDim.x * blockDim.x) {
        const float v = x[idx];
        const int c = (int)(idx % C);
        atomicAdd(&sbuf[c], v);
        atomicAdd(&sbuf[C + c], v * v);
    }
    __syncthreads();
    for (int i = threadIdx.x; i < 2 * C; i += blockDim.x)
        atomicAdd(&sums[i], sbuf[i]);
}

__device__ __forceinline__ float dice_elem(float x, float m, float v,
                                           float gamma, float bbeta,
                                           float alpha, float beta) {
    const float bn = (x - m) * rsqrtf(v + 1e-8f) * gamma + bbeta;
    const float xn = 1.f / (1.f + __expf(-beta * bn));
    return alpha * (1.f - xn) * x + xn * x;
}

// dice3 over X1 (R x 36) fused with the 36->1 GEMV (au_W2, au_b2) -> AU[r]
__global__ __launch_bounds__(256) void dice3_auout(
    const float* __restrict__ X1, const float* __restrict__ S36,
    const float* __restrict__ alpha, const float* __restrict__ beta,
    const float* __restrict__ gamma, const float* __restrict__ bbeta,
    const float* __restrict__ W2, const float* __restrict__ b2,
    float* __restrict__ AU, int R) {
    const int r = blockIdx.x * blockDim.x + threadIdx.x;
    if (r >= R) return;
    const float invN = 1.f / (float)(BATCH * TOPK);
    float acc = b2[0];
#pragma unroll 4
    for (int c = 0; c < 36; ++c) {
        const float m = S36[c] * invN;
        const float v = fmaxf(S36[36 + c] * invN - m * m, 0.f);
        const float x = X1[(size_t)r * 36 + c];
        acc += dice_elem(x, m, v, gamma[c], bbeta[c], alpha[c], beta[c]) * W2[c];
    }
    AU[r] = acc;
}

// res[b][0..63] = item_cat, res[b][64..159] = sum_k topk_full * au_out
__global__ __launch_bounds__(256) void build_res(
    const float* __restrict__ X, const float* __restrict__ TR,
    const float* __restrict__ AU, float* __restrict__ RES) {
    const int t = blockIdx.x * blockDim.x + threadIdx.x;
    if (t >= BATCH * 160) return;
    const int b = t / 160, d = t % 160;
    float out;
    if (d < 64) {
        out = X[(size_t)(b * TOPK) * 132 + 65 + d];
    } else {
        const int j = d - 64;
        float acc = 0.f;
#pragma unroll
        for (int k = 0; k < TOPK; ++k) {
            const int row = b * TOPK + k;
            const float tf = (j < 64) ? X[(size_t)row * 132 + j]
                                      : TR[(size_t)row * HD + (j - 64)];
            acc += tf * AU[row];
        }
        out = acc;
    }
    RES[(size_t)b * 160 + d] = out;
}

// elementwise dice2 over R x C
__global__ __launch_bounds__(256) void dice2_apply(
    const float* __restrict__ Hin, const float* __restrict__ S,
    const float* __restrict__ alpha, const float* __restrict__ beta,
    const float* __restrict__ gamma, const float* __restrict__ bbeta,
    float* __restrict__ Hout, int R, int C) {
    const size_t idx = (size_t)blockIdx.x * blockDim.x + threadIdx.x;
    if (idx >= (size_t)R * C) return;
    const int c = (int)(idx % C);
    const float invN = 1.f / (float)R;
    const float m = S[c] * invN;
    const float v = fmaxf(S[C + c] * invN - m * m, 0.f);
    Hout[idx] = dice_elem(Hin[idx], m, v, gamma[c], bbeta[c], alpha[c], beta[c]);
}

// dice2 over H2 (B x 40) fused with 40->1 GEMV, sigmoid, per-block loss partials
__global__ __launch_bounds__(256) void final_kernel(
    const float* __restrict__ H2, const float* __restrict__ S40,
    const float* __restrict__ alpha, const float* __restrict__ beta,
    const float* __restrict__ gamma, const float* __restrict__ bbeta,
    const float* __restrict__ W3, const float* __restrict__ b3,
    const int* __restrict__ lb, float* __restrict__ probs,
    float* __restrict__ lpart) {
    __shared__ float red[256];
    const int b = blockIdx.x * 256 + threadIdx.x;
    const float invN = 1.f / (float)BATCH;
    float lg = b3[0];
#pragma unroll 4
    for (int c = 0; c < 40; ++c) {
        const float m = S40[c] * invN;
        const float v = fmaxf(S40[40 + c] * invN - m * m, 0.f);
        const float x = H2[(size_t)b * 40 + c];
        lg += dice_elem(x, m, v, gamma[c], bbeta[c], alpha[c], beta[c]) * W3[c];
    }
    probs[b] = 1.f / (1.f + expf(-lg));
    const float lab = (float)lb[b];
    red[threadIdx.x] = fmaxf(lg, 0.f) - lg * lab + log1pf(expf(-fabsf(lg)));
    __syncthreads();
    for (int off = 128; off > 0; off >>= 1) {
        if (threadIdx.x < off) red[threadIdx.x] += red[threadIdx.x + off];
        __syncthreads();
    }
    if (threadIdx.x == 0) lpart[blockIdx.x] = red[0];
}

__global__ void loss_fin(const float* __restrict__ lpart, float* __restrict__ out) {
    __shared__ float s[64];
    s[threadIdx.x] = lpart[threadIdx.x];
    __syncthreads();
    for (int off = 32; off > 0; off >>= 1) {
        if (threadIdx.x < off) s[threadIdx.x] += s[threadIdx.x + off];
        __syncthreads();
    }
    if (threadIdx.x == 0) out[BATCH] = s[0] / (float)BATCH;
}

// ---------------------------------------------------------------------------
extern "C" void kernel_launch(void* const* d_in, const int* in_sizes, int n_in,
                              void* d_out, int out_size, void* d_ws, size_t ws_size,
                              hipStream_t stream) {
    const int*   iid       = (const int*)d_in[0];
    const int*   aid       = (const int*)d_in[1];
    const int*   lb        = (const int*)d_in[2];
    const int*   hist_iid  = (const int*)d_in[3];
    const int*   hist_aid  = (const int*)d_in[4];
    const int*   hist_rate = (const int*)d_in[5];
    const float* item_tab  = (const float*)d_in[6];
    const float* cate_tab  = (const float*)d_in[7];
    const float* rate_tab  = (const float*)d_in[8];
    const float* au_W1     = (const float*)d_in[9];
    const float* au_b1     = (const float*)d_in[10];
    const float* au_alpha  = (const float*)d_in[11];
    const float* au_beta   = (const float*)d_in[12];
    const float* au_gamma  = (const float*)d_in[13];
    const float* au_bbeta  = (const float*)d_in[14];
    const float* au_W2     = (const float*)d_in[15];
    const float* au_b2     = (const float*)d_in[16];
    const float* l1_W      = (const float*)d_in[17];
    const float* l1_b      = (const float*)d_in[18];
    const float* d1_alpha  = (const float*)d_in[19];
    const float* d1_beta   = (const float*)d_in[20];
    const float* d1_gamma  = (const float*)d_in[21];
    const float* d1_bbeta  = (const float*)d_in[22];
    const float* l2_W      = (const float*)d_in[23];
    const float* l2_b      = (const float*)d_in[24];
    const float* d2_alpha  = (const float*)d_in[25];
    const float* d2_beta   = (const float*)d_in[26];
    const float* d2_gamma  = (const float*)d_in[27];
    const float* d2_bbeta  = (const float*)d_in[28];
    const float* l3_W      = (const float*)d_in[29];
    const float* l3_b      = (const float*)d_in[30];
    float* out = (float*)d_out;

    // workspace layout (floats)
    float* ws = (float*)d_ws;
    size_t off = 0;
    float* X   = ws + off; off += (size_t)BATCH * TOPK * 132;  // attn inputs, padded K
    float* X1  = ws + off; off += (size_t)BATCH * TOPK * 36;   // X @ au_W1 + b1
    float* TR  = ws + off; off += (size_t)BATCH * TOPK * HD;   // topk rate emb
    float* AU  = ws + off; off += (size_t)BATCH * TOPK;        // attn scores
    float* RES = ws + off; off += (size_t)BATCH * 160;
    float* H1  = ws + off; off += (size_t)BATCH * 80;
    float* H1D = ws + off; off += (size_t)BATCH * 80;
    float* H2  = ws + off; off += (size_t)BATCH * 40;
    float* W1P = ws + off; off += 132 * 48;                    // padded au_W1
    float* W2P = ws + off; off += 80 * 48;                     // padded l2_W
    float* S36 = ws + off; off += 72;                          // sum/sumsq (36 ch)
    float* S80 = ws + off; off += 160;
    float* S40 = ws + off; off += 80;
    float* LP  = ws + off; off += 64;                          // loss partials

    // zero the reduction accumulators (capture-safe)
    hipMemsetAsync(S36, 0, (72 + 160 + 80) * sizeof(float), stream);

    prep_weights<<<25, 256, 0, stream>>>(au_W1, l2_W, W1P, W2P);

    build_x<<<BATCH / 8, 256, 0, stream>>>(iid, aid, hist_iid, hist_aid, hist_rate,
                                           item_tab, cate_tab, rate_tab, X, TR);

    // X1 = X @ W1P + au_b1   (M=81920, K=132, Nlog=36, 3 n-tiles)
    wmma_gemm_f32<3><<<BATCH * TOPK / 16 / 8, 256, 0, stream>>>(
        X, 132, W1P, 48, au_b1, X1, 36, 36, 132);

    stats_kernel<<<512, 256, 0, stream>>>(X1, BATCH * TOPK, 36, S36);

    dice3_auout<<<(BATCH * TOPK + 255) / 256, 256, 0, stream>>>(
        X1, S36, au_alpha, au_beta, au_gamma, au_bbeta, au_W2, au_b2,
        AU, BATCH * TOPK);

    build_res<<<(BATCH * 160 + 255) / 256, 256, 0, stream>>>(X, TR, AU, RES);

    // H1 = RES @ l1_W + l1_b   (M=16384, K=160, N=80, 5 n-tiles)
    wmma_gemm_f32<5><<<BATCH / 16 / 8, 256, 0, stream>>>(
        RES, 160, l1_W, 80, l1_b, H1, 80, 80, 160);

    stats_kernel<<<512, 256, 0, stream>>>(H1, BATCH, 80, S80);

    dice2_apply<<<(BATCH * 80 + 255) / 256, 256, 0, stream>>>(
        H1, S80, d1_alpha, d1_beta, d1_gamma, d1_bbeta, H1D, BATCH, 80);

    // H2 = H1D @ W2P + l2_b   (M=16384, K=80, Nlog=40, 3 n-tiles)
    wmma_gemm_f32<3><<<BATCH / 16 / 8, 256, 0, stream>>>(
        H1D, 80, W2P, 48, l2_b, H2, 40, 40, 80);

    stats_kernel<<<512, 256, 0, stream>>>(H2, BATCH, 40, S40);

    final_kernel<<<BATCH / 256, 256, 0, stream>>>(
        H2, S40, d2_alpha, d2_beta, d2_gamma, d2_bbeta, l3_W, l3_b,
        lb, out, LP);

    loss_fin<<<1, 64, 0, stream>>>(LP, out);
}